// MSDeformableAttention_10763188044230
// MI455X (gfx1250) — compile-verified
//
#include <hip/hip_runtime.h>
#include <cstdint>

typedef __attribute__((ext_vector_type(2))) float v2f;
typedef __attribute__((ext_vector_type(8))) float v8f;

#define NHh   8
#define HDd   32
#define NLl   3
#define NPp   4
#define Bb    4
#define NQq   300
#define NKk   17
#define LQq   (NQq * NKk)   /* 5100 */
#define LVv   13125         /* 100*100 + 50*50 + 25*25 */
#define EMB   256
#define KVAL  512

// ---------------------------------------------------------------------------
// Generic GEMM: C[M,N] = A[M,K] @ W[N,K]^T + bias[N], optional row mask->0.
// Full-precision fp32 matrix path via V_WMMA_F32_16X16X4_F32 (wave32).
// Each wave computes a 16(M) x (16*NT)(N) strip: the A fragment is loaded
// once per k-step and reused across NT WMMAs (A traffic / NT, WMMA density up).
// 8 waves / 256-thread block.
// Fragment layouts per CDNA5 ISA 7.12.2:
//   A 16x4 : lane&15 = M-row, lane>>4 selects K pair {0,1} vs {2,3}
//   B 4x16 : lane&15 = N-col, lane>>4 selects K pair
//   C 16x16: VGPR r -> row r + 8*(lane>>4), col = lane&15
// Requires (N/16) % NT == 0 (true for N = 256, 384 with NT=4; N = 96 with NT=2).
// ---------------------------------------------------------------------------
template <int NT>
__global__ __launch_bounds__(256)
void wmma_gemm_nt(const float* __restrict__ A,
                  const float* __restrict__ W,
                  const float* __restrict__ bias,
                  const unsigned char* __restrict__ row_mask,
                  float* __restrict__ C,
                  int M, int N, int K)
{
    const int lane = threadIdx.x & 31;
    const int wave = threadIdx.x >> 5;

    const long groups_n = (long)(N >> 4) / NT;       // N-strips per row of tiles
    const long tiles_m  = (long)(M + 15) >> 4;
    const long group    = (long)blockIdx.x * 8 + wave;
    if (group >= tiles_m * groups_n) return;         // wave-uniform: EXEC all-1s

    const int tm = (int)(group / groups_n);
    const int gn = (int)(group % groups_n);
    const int m0 = tm << 4;
    const int n0 = gn * (16 * NT);

    const int  mrow    = m0 + (lane & 15);
    const bool a_valid = (mrow < M);
    const int  kpair   = (lane >> 4) << 1;           // 0 or 2

    const float* __restrict__ Arow = A + (long)(a_valid ? mrow : 0) * K + kpair;
    const float* __restrict__ Wrow = W + (long)(n0 + (lane & 15)) * K + kpair;

    v8f acc[NT];
    #pragma unroll
    for (int t = 0; t < NT; ++t) acc[t] = (v8f){0.f,0.f,0.f,0.f,0.f,0.f,0.f,0.f};

    #pragma unroll 4
    for (int k = 0; k < K; k += 4) {
        v2f a = *(const v2f*)(Arow + k);             // A[m][k+kpair .. +1]
        if (!a_valid) { a.x = 0.f; a.y = 0.f; }
        #pragma unroll
        for (int t = 0; t < NT; ++t) {
            // W[n0 + t*16 + (lane&15)][k+kpair .. +1] == B[k][n]
            v2f b = *(const v2f*)(Wrow + (long)t * 16 * K + k);
            acc[t] = __builtin_amdgcn_wmma_f32_16x16x4_f32(
                         /*neg_a=*/false, a, /*neg_b=*/false, b,
                         /*c_mod=*/(short)0, acc[t],
                         /*reuse_a=*/false, /*reuse_b=*/false);
        }
    }

    const int rbase = m0 + ((lane >> 4) << 3);

    #pragma unroll
    for (int t = 0; t < NT; ++t) {
        const int   col = n0 + t * 16 + (lane & 15);
        const float bv  = bias ? bias[col] : 0.f;
        #pragma unroll
        for (int r = 0; r < 8; ++r) {
            const int row = rbase + r;
            if (row < M) {
                float v = acc[t][r] + bv;
                if (row_mask && row_mask[row]) v = 0.f;
                C[(long)row * N + col] = v;
            }
        }
    }
}

// ---------------------------------------------------------------------------
// Softmax over the trailing 12 (= NL*NP) attention logits per (b, q, head).
// aw layout [b, lq, NH, 12] is contiguous, so group i starts at i*12.
// ---------------------------------------------------------------------------
__global__ __launch_bounds__(256)
void softmax12_kernel(float* __restrict__ aw, int total)
{
    const int i = blockIdx.x * blockDim.x + threadIdx.x;
    if (i >= total) return;
    float* p = aw + (long)i * 12;
    float m = p[0];
    #pragma unroll
    for (int j = 1; j < 12; ++j) m = fmaxf(m, p[j]);
    float e[12];
    float s = 0.f;
    #pragma unroll
    for (int j = 0; j < 12; ++j) { e[j] = __expf(p[j] - m); s += e[j]; }
    const float inv = 1.f / s;
    #pragma unroll
    for (int j = 0; j < 12; ++j) p[j] = e[j] * inv;
}

// ---------------------------------------------------------------------------
// Deformable sampling. One wave per (b, q, head); lane = channel (HD==32).
// All location/weight math is wave-uniform -> no divergence around gathers;
// each corner gather is one coalesced 128B line. v_proj (53.8 MB) fits in the
// 192 MB L2, so the 96 gathers/wave are L2-resident.
// ---------------------------------------------------------------------------
__global__ __launch_bounds__(256)
void msda_sample_kernel(const float* __restrict__ vproj, // [b, LV, 256]
                        const float* __restrict__ off,   // [b, lq, NH, NL, NP, 4]
                        const float* __restrict__ aw,    // [b, lq, NH, NL*NP] softmaxed
                        const float* __restrict__ refl,  // [b, 300, 3, 17, 2]
                        const float* __restrict__ refr,  // [b, 300, 3, 17, 2]
                        float* __restrict__ outp)        // [b, lq, 256]
{
    const int lane = threadIdx.x & 31;
    const int g = (int)((blockIdx.x * (long)blockDim.x + threadIdx.x) >> 5);
    if (g >= Bb * LQq * NHh) return;

    const int h  = g % NHh;
    const int bq = g / NHh;
    const int q  = bq % LQq;
    const int b  = bq / LQq;
    const int nq_i = q / NKk;
    const int nk_i = q % NKk;

    const float* __restrict__ offp = off + (long)bq * (NHh * NLl * NPp * 4) + h * (NLl * NPp * 4);
    const float* __restrict__ awp  = aw  + (long)bq * (NHh * NLl * NPp)     + h * (NLl * NPp);

    constexpr int Hs[3]     = {100, 50, 25};
    constexpr int Ws[3]     = {100, 50, 25};
    constexpr int starts[3] = {0, 10000, 12500};

    float acc = 0.f;

    #pragma unroll
    for (int l = 0; l < NLl; ++l) {
        const long rbase = ((((long)b * NQq + nq_i) * NLl + l) * NKk + nk_i) * 2;
        const float rxl = refl[rbase + 0];
        const float ryl = refl[rbase + 1];
        const float rxr = refr[rbase + 0];
        const float ryr = refr[rbase + 1];

        const int Hh = Hs[l], Ww = Ws[l];
        const float invW = 1.f / (float)Ww;
        const float invH = 1.f / (float)Hh;
        const float* __restrict__ vlev =
            vproj + ((long)b * LVv + starts[l]) * EMB + h * HDd + lane;

        #pragma unroll
        for (int p = 0; p < 8; ++p) {
            const int  pp    = p & 3;
            const bool right = (p >= 4);
            const float ox = offp[(l * NPp + pp) * 4 + (right ? 2 : 0)];
            const float oy = offp[(l * NPp + pp) * 4 + (right ? 3 : 1)];
            const float lx = (right ? rxr : rxl) + ox * invW;
            const float ly = (right ? ryr : ryl) + oy * invH;
            const float w  = awp[l * NPp + pp];

            const float x  = lx * (float)Ww - 0.5f;
            const float y  = ly * (float)Hh - 0.5f;
            const float xf = floorf(x);
            const float yf = floorf(y);
            const float tx = x - xf;
            const float ty = y - yf;
            const int x0 = (int)xf;
            const int y0 = (int)yf;

            #pragma unroll
            for (int c = 0; c < 4; ++c) {
                const int dx = c & 1;
                const int dy = c >> 1;
                const int xi = x0 + dx;
                const int yi = y0 + dy;
                if (xi >= 0 && xi < Ww && yi >= 0 && yi < Hh) {   // wave-uniform
                    const float wc = (dx ? tx : 1.f - tx) * (dy ? ty : 1.f - ty);
                    const float gv = vlev[(long)(yi * Ww + xi) * EMB];
                    acc = fmaf(w * wc, gv, acc);
                }
            }
        }
    }

    outp[(long)bq * EMB + h * HDd + lane] = acc;
}

// ---------------------------------------------------------------------------
// Host-side launch.
// ---------------------------------------------------------------------------
extern "C" void kernel_launch(void* const* d_in, const int* in_sizes, int n_in,
                              void* d_out, int out_size, void* d_ws, size_t ws_size,
                              hipStream_t stream)
{
    (void)in_sizes; (void)n_in; (void)out_size; (void)ws_size;

    const float* query = (const float*)d_in[0];   // [4, 5100, 256]
    const float* refl  = (const float*)d_in[1];   // [4, 300, 3, 17, 2]
    const float* refr  = (const float*)d_in[2];   // [4, 300, 3, 17, 2]
    const float* value = (const float*)d_in[3];   // [4, 13125, 512]
    /* d_in[4] = value_spatial_shapes (static, hardcoded) */
    const unsigned char* vmask = (const unsigned char*)d_in[5];  // [4, 13125] bool
    const float* W_off  = (const float*)d_in[6];  // [384, 256]
    const float* b_off  = (const float*)d_in[7];
    const float* W_attn = (const float*)d_in[8];  // [96, 256]
    const float* b_attn = (const float*)d_in[9];
    const float* W_val  = (const float*)d_in[10]; // [256, 512]
    const float* b_val  = (const float*)d_in[11];
    const float* W_out  = (const float*)d_in[12]; // [256, 256]
    const float* b_out  = (const float*)d_in[13];
    float* out = (float*)d_out;                   // [4, 5100, 256]

    const int Mv = Bb * LVv;   // 52500
    const int Mq = Bb * LQq;   // 20400

    // workspace carve-up (bytes)
    char* ws = (char*)d_ws;
    float* v_proj = (float*)(ws);                                   // 52500*256*4 = 53,760,000
    float* offbuf = (float*)(ws + 53760000);                        // 20400*384*4 = 31,334,400
    float* awbuf  = (float*)(ws + 53760000 + 31334400);             // 20400*96*4  =  7,833,600
    float* prebuf = (float*)(ws + 53760000 + 31334400 + 7833600);   // 20400*256*4 = 20,889,600

    auto gemm_blocks = [](int M, int N, int NT) -> int {
        long groups = (long)((M + 15) / 16) * ((N / 16) / NT);
        return (int)((groups + 7) / 8);
    };

    // 1) value projection + mask:  v = value @ W_val^T + b_val  (masked rows -> 0)
    wmma_gemm_nt<4><<<gemm_blocks(Mv, EMB, 4), 256, 0, stream>>>(
        value, W_val, b_val, vmask, v_proj, Mv, EMB, KVAL);

    // 2) sampling offsets: off = query @ W_off^T + b_off
    wmma_gemm_nt<4><<<gemm_blocks(Mq, 384, 4), 256, 0, stream>>>(
        query, W_off, b_off, nullptr, offbuf, Mq, 384, EMB);

    // 3) attention logits: aw = query @ W_attn^T + b_attn
    wmma_gemm_nt<2><<<gemm_blocks(Mq, 96, 2), 256, 0, stream>>>(
        query, W_attn, b_attn, nullptr, awbuf, Mq, 96, EMB);

    // 4) softmax over NL*NP = 12 per (b, q, head)
    {
        const int total = Bb * LQq * NHh;  // 163200
        softmax12_kernel<<<(total + 255) / 256, 256, 0, stream>>>(awbuf, total);
    }

    // 5) deformable bilinear sampling: one wave per (b, q, head)
    {
        const long waves = (long)Bb * LQq * NHh;   // 163200
        const int blocks = (int)((waves * 32 + 255) / 256);
        msda_sample_kernel<<<blocks, 256, 0, stream>>>(
            v_proj, offbuf, awbuf, refl, refr, prebuf);
    }

    // 6) output projection: out = pre @ W_out^T + b_out
    wmma_gemm_nt<4><<<gemm_blocks(Mq, EMB, 4), 256, 0, stream>>>(
        prebuf, W_out, b_out, nullptr, out, Mq, EMB, EMB);
}